// GNN_DQN_83966610637551
// MI455X (gfx1250) — compile-verified
//
#include <hip/hip_runtime.h>
#include <hip/hip_bf16.h>
#include <stdint.h>

typedef float v2f __attribute__((ext_vector_type(2)));
typedef float v8f __attribute__((ext_vector_type(8)));

#define FIN  128
#define HID  256
#define NACT 8

// ---------- CSR build ----------------------------------------------------

__global__ void k_zero_i32(int* __restrict__ p, int n) {
  int i = blockIdx.x * blockDim.x + threadIdx.x;
  if (i < n) p[i] = 0;
}

__global__ void k_hist(const long long* __restrict__ dst, int* __restrict__ cnt, int E) {
  int e = blockIdx.x * blockDim.x + threadIdx.x;
  if (e < E) atomicAdd(&cnt[(int)dst[e]], 1);
}

// Single block: chunked exclusive scan of cnt -> rowStart, plus
// dinv[n] = rsqrt(deg_in + 1 self-loop) and cursor reset.
__global__ void k_scan_dinv(const int* __restrict__ cnt, int* __restrict__ rowStart,
                            int* __restrict__ cursor, float* __restrict__ dinv, int N) {
  __shared__ int sums[256];
  int t = threadIdx.x;
  int ch = (N + 255) >> 8;
  int lo = t * ch;
  int hi = min(lo + ch, N);
  int s = 0;
  for (int n = lo; n < hi; ++n) s += cnt[n];
  sums[t] = s;
  __syncthreads();
  for (int off = 1; off < 256; off <<= 1) {
    int v = (t >= off) ? sums[t - off] : 0;
    __syncthreads();
    sums[t] += v;
    __syncthreads();
  }
  int run = (t > 0) ? sums[t - 1] : 0;
  for (int n = lo; n < hi; ++n) {
    rowStart[n] = run;
    run += cnt[n];
    dinv[n] = rsqrtf((float)(cnt[n] + 1));  // +1 for self-loop
    cursor[n] = 0;
  }
  if (t == 255) rowStart[N] = run;  // == E
}

__global__ void k_fill(const long long* __restrict__ src, const long long* __restrict__ dst,
                       const int* __restrict__ rowStart, int* __restrict__ cursor,
                       int* __restrict__ csrSrc, int E) {
  int e = blockIdx.x * blockDim.x + threadIdx.x;
  if (e < E) {
    int d = (int)dst[e];
    int p = rowStart[d] + atomicAdd(&cursor[d], 1);
    csrSrc[p] = (int)src[e];
  }
}

// Sort each row's src list (insertion sort, avg deg ~32) so the f32
// summation order in aggregation is deterministic across calls.
__global__ void k_sort_rows(const int* __restrict__ rowStart, int* __restrict__ csrSrc, int N) {
  int n = blockIdx.x * blockDim.x + threadIdx.x;
  if (n >= N) return;
  int s0 = rowStart[n], s1 = rowStart[n + 1];
  for (int i = s0 + 1; i < s1; ++i) {
    int key = csrSrc[i];
    int j = i - 1;
    while (j >= s0 && csrSrc[j] > key) { csrSrc[j + 1] = csrSrc[j]; --j; }
    csrSrc[j + 1] = key;
  }
}

// ---------- f32 WMMA GEMM: C[M x 256] = A[M x K] @ B[K x 256] ------------
// One wave32 per 16x16 output tile; V_WMMA_F32_16X16X4_F32 K-loop.
// A fragment (16x4): lanes 0-15 rows, VGPR0/1 = K pair, lane half selects K+0/K+2.
// B fragment (4x16): lanes = N cols, VGPR0/1 = K row pair, lane half selects K+0/K+2.

template <int K, bool BIAS_RELU>
__global__ __launch_bounds__(256) void k_gemm(const float* __restrict__ A,
                                              const float* __restrict__ B,
                                              const float* __restrict__ bias,
                                              float* __restrict__ C, int M) {
  int wave  = blockIdx.x * 8 + (threadIdx.x >> 5);
  int lane  = threadIdx.x & 31;
  int mTile = wave >> 4;   // 16 N-tiles (256 cols) per M-strip
  int nTile = wave & 15;
  if (mTile * 16 >= M) return;  // wave-uniform; EXEC stays all-ones
  int m0 = mTile * 16, n0 = nTile * 16;
  int half2 = (lane >> 4) * 2;  // 0 or 2 : K offset for this lane half
  int l15   = lane & 15;

  const float* Ap = A + (size_t)(m0 + l15) * K + half2;
  const float* Bp = B + (size_t)half2 * HID + n0 + l15;

  v8f acc = {};
#pragma unroll 8
  for (int k = 0; k < K; k += 4) {
    v2f a;
    a[0] = Ap[k];
    a[1] = Ap[k + 1];
    v2f b;
    b[0] = Bp[(size_t)k * HID];
    b[1] = Bp[(size_t)k * HID + HID];
    acc = __builtin_amdgcn_wmma_f32_16x16x4_f32(false, a, false, b, (short)0, acc,
                                                false, false);
  }

  int col = n0 + l15;
  int r0  = m0 + (lane >> 4) * 8;
  float bv = BIAS_RELU ? bias[col] : 0.0f;
#pragma unroll
  for (int r = 0; r < 8; ++r) {
    float v = acc[r];
    if (BIAS_RELU) v = fmaxf(v + bv, 0.0f);
    C[(size_t)(r0 + r) * HID + col] = v;
  }
}

// ---------- GCN aggregation: out[n] = relu(sum norm*h[src] + b) ----------
// One 64-thread block per node, 4 floats (b128) per thread; pure gathers,
// everything L2-resident, no atomics.

__global__ __launch_bounds__(64) void k_aggregate(const float* __restrict__ h,
                                                  const float* __restrict__ dinv,
                                                  const int* __restrict__ rowStart,
                                                  const int* __restrict__ csrSrc,
                                                  const float* __restrict__ bias,
                                                  float* __restrict__ out) {
  int n = blockIdx.x;
  int f = threadIdx.x * 4;
  float dn = dinv[n];

  float4 hv = *(const float4*)(h + (size_t)n * HID + f);
  float ws = dn * dn;  // self-loop norm
  float ax = ws * hv.x, ay = ws * hv.y, az = ws * hv.z, aw = ws * hv.w;

  int s0 = rowStart[n], s1 = rowStart[n + 1];
  for (int j = s0; j < s1; ++j) {
    int s = csrSrc[j];
    float w = dn * dinv[s];
    float4 v = *(const float4*)(h + (size_t)s * HID + f);
    ax = fmaf(w, v.x, ax);
    ay = fmaf(w, v.y, ay);
    az = fmaf(w, v.z, az);
    aw = fmaf(w, v.w, aw);
  }
  float4 bv = *(const float4*)(bias + f);
  float4 o;
  o.x = fmaxf(ax + bv.x, 0.0f);
  o.y = fmaxf(ay + bv.y, 0.0f);
  o.z = fmaxf(az + bv.z, 0.0f);
  o.w = fmaxf(aw + bv.w, 0.0f);
  *(float4*)(out + (size_t)n * HID + f) = o;
}

// ---------- head: Q[N x 8] = H[N x 256] @ Wh2[256 x 8] + bh2 -------------

__global__ __launch_bounds__(256) void k_head(const float* __restrict__ H,
                                              const float* __restrict__ W,
                                              const float* __restrict__ bias,
                                              float* __restrict__ Q, int N) {
  __shared__ float w[HID * NACT];
  int t = threadIdx.x;
  for (int i = t; i < HID * NACT; i += 256) w[i] = W[i];
  __syncthreads();
  int n = blockIdx.x * 32 + (t >> 3);
  int a = t & 7;
  if (n >= N) return;
  const float4* hr = (const float4*)(H + (size_t)n * HID);
  float acc = bias[a];
#pragma unroll 4
  for (int k4 = 0; k4 < HID / 4; ++k4) {
    float4 hv = hr[k4];
    int kb = k4 * 4 * NACT + a;
    acc = fmaf(hv.x, w[kb], acc);
    acc = fmaf(hv.y, w[kb + NACT], acc);
    acc = fmaf(hv.z, w[kb + 2 * NACT], acc);
    acc = fmaf(hv.w, w[kb + 3 * NACT], acc);
  }
  Q[(size_t)n * NACT + a] = acc;
}

// ---------- launch -------------------------------------------------------

extern "C" void kernel_launch(void* const* d_in, const int* in_sizes, int n_in,
                              void* d_out, int out_size, void* d_ws, size_t ws_size,
                              hipStream_t stream) {
  const float*     x   = (const float*)d_in[0];
  const long long* ei  = (const long long*)d_in[1];
  const float*     W1  = (const float*)d_in[2];
  const float*     b1  = (const float*)d_in[3];
  const float*     W2  = (const float*)d_in[4];
  const float*     b2  = (const float*)d_in[5];
  const float*     Wh1 = (const float*)d_in[6];
  const float*     bh1 = (const float*)d_in[7];
  const float*     Wh2 = (const float*)d_in[8];
  const float*     bh2 = (const float*)d_in[9];

  int N = in_sizes[0] / FIN;   // 10000
  int E = in_sizes[1] / 2;     // 320000
  const long long* src = ei;
  const long long* dst = ei + E;

  uintptr_t p = (uintptr_t)d_ws;
  auto carve = [&](size_t bytes) -> uintptr_t {
    uintptr_t r = p;
    p += (bytes + 255) & ~(size_t)255;
    return r;
  };
  float* dinv     = (float*)carve((size_t)N * 4);
  int*   cnt      = (int*)carve((size_t)N * 4);
  int*   rowStart = (int*)carve((size_t)(N + 1) * 4);
  int*   cursor   = (int*)carve((size_t)N * 4);
  int*   csrSrc   = (int*)carve((size_t)E * 4);
  float* bufA     = (float*)carve((size_t)N * HID * 4);
  float* bufB     = (float*)carve((size_t)N * HID * 4);
  (void)ws_size; (void)n_in; (void)out_size;

  dim3 b256(256);
  // CSR build (shared by both GCN layers)
  k_zero_i32<<<(N + 255) / 256, b256, 0, stream>>>(cnt, N);
  k_hist<<<(E + 255) / 256, b256, 0, stream>>>(dst, cnt, E);
  k_scan_dinv<<<1, b256, 0, stream>>>(cnt, rowStart, cursor, dinv, N);
  k_fill<<<(E + 255) / 256, b256, 0, stream>>>(src, dst, rowStart, cursor, csrSrc, E);
  k_sort_rows<<<(N + 255) / 256, b256, 0, stream>>>(rowStart, csrSrc, N);

  int mTiles = (N + 15) / 16;                  // 625
  int gemmBlocks = (mTiles * 16 + 7) / 8;      // 8 waves/block, 16 n-tiles/m-strip

  // Layer 1: h1 = x@W1 ; out1 = relu(agg(h1) + b1)
  k_gemm<FIN, false><<<gemmBlocks, b256, 0, stream>>>(x, W1, nullptr, bufA, N);
  k_aggregate<<<N, 64, 0, stream>>>(bufA, dinv, rowStart, csrSrc, b1, bufB);
  // Layer 2: h2 = out1@W2 ; out2 = relu(agg(h2) + b2)
  k_gemm<HID, false><<<gemmBlocks, b256, 0, stream>>>(bufB, W2, nullptr, bufA, N);
  k_aggregate<<<N, 64, 0, stream>>>(bufA, dinv, rowStart, csrSrc, b2, bufB);
  // MLP: h3 = relu(out2@Wh1 + bh1)
  k_gemm<HID, true><<<gemmBlocks, b256, 0, stream>>>(bufB, Wh1, bh1, bufA, N);
  // Head: q = h3@Wh2 + bh2
  k_head<<<(N + 31) / 32, b256, 0, stream>>>(bufA, Wh2, bh2, (float*)d_out, N);
}